// ResFreq_28724741276003
// MI455X (gfx1250) — compile-verified
//
#include <hip/hip_runtime.h>

// ---------------------------------------------------------------------------
// ResFreq: 131072 batched 8x8 symmetric eigensolves + transition postprocess.
// Phase 1: one thread = one matrix, cyclic Jacobi in registers, rank-sorted
//          eigenpairs handed to LDS.
// Phase 2: wave-cooperative; vu/vv computed as evec x selection-matrix GEMMs
//          on V_WMMA_F32_16X16X4_F32 (two matrices block-packed per 16x16
//          tile), outputs stored with wide/coalesced stores.
// ---------------------------------------------------------------------------

typedef __attribute__((ext_vector_type(2))) float v2f;
typedef __attribute__((ext_vector_type(4))) float f4;
typedef __attribute__((ext_vector_type(8))) float v8f;

#define SDIM    8
#define NPAIR   28
#define NMAT    (32 * 64 * 64)   // 131072 matrices
#define MPB     64               // matrices (threads) per block
#define NSWEEP  8                // fixed Jacobi sweep count (deterministic)

// triu_indices(8, k=1) row-major: (i,j) from flat pair index k.
__device__ __forceinline__ void pair_ij(int k, int& i, int& j) {
  const int offs[8] = {0, 7, 13, 18, 22, 25, 27, 28};
  int ii = 0, off = 0;
#pragma unroll
  for (int t = 0; t < 7; ++t) {
    if (k >= offs[t]) { ii = t; off = offs[t]; }
  }
  i = ii;
  j = ii + 1 + (k - off);
}

__global__ __launch_bounds__(MPB) void resfreq_kernel(
    const float* __restrict__ F, const float* __restrict__ Gz,
    const float* __restrict__ fl, const float* __restrict__ fh,
    const float* __restrict__ resf, float* __restrict__ out) {
  // LDS handoff: sorted eigenvalues + eigenvector matrices, per local matrix.
  __shared__ float evS[MPB * 8];        // [slot][sorted l]
  __shared__ float evecS[MPB * 64];     // [slot][s][sorted l]

  const int tid = threadIdx.x;
  const int g   = blockIdx.x * MPB + tid;   // this thread's matrix id
  const float rf = resf[0];

  // gfx1250 prefetch of the per-block frequency windows (global_prefetch_b8).
  __builtin_prefetch(&fl[blockIdx.x * MPB], 0, 0);
  __builtin_prefetch(&fh[blockIdx.x * MPB], 0, 0);

  // ------------------------- Phase 1: Jacobi ------------------------------
  float H[8][8];
  float V[8][8];
  {
    const float* Fb = F + (size_t)g * 64;
    const float* Gb = Gz + (size_t)g * 64;
#pragma unroll
    for (int r = 0; r < 8; ++r) {
      f4 fa = *(const f4*)(Fb + r * 8);
      f4 fb = *(const f4*)(Fb + r * 8 + 4);
      f4 ga = *(const f4*)(Gb + r * 8);
      f4 gb = *(const f4*)(Gb + r * 8 + 4);
      H[r][0] = fmaf(rf, ga.x, fa.x);
      H[r][1] = fmaf(rf, ga.y, fa.y);
      H[r][2] = fmaf(rf, ga.z, fa.z);
      H[r][3] = fmaf(rf, ga.w, fa.w);
      H[r][4] = fmaf(rf, gb.x, fb.x);
      H[r][5] = fmaf(rf, gb.y, fb.y);
      H[r][6] = fmaf(rf, gb.z, fb.z);
      H[r][7] = fmaf(rf, gb.w, fb.w);
#pragma unroll
      for (int c2 = 0; c2 < 8; ++c2) V[r][c2] = (r == c2) ? 1.f : 0.f;
    }
  }

#pragma unroll 1
  for (int sweep = 0; sweep < NSWEEP; ++sweep) {
#pragma unroll
    for (int p = 0; p < 7; ++p) {
#pragma unroll
      for (int q = p + 1; q < 8; ++q) {
        const float apq = H[p][q];
        const float d   = H[q][q] - H[p][p];
        const float r   = sqrtf(d * d + 4.f * apq * apq);
        const float den = d + ((d >= 0.f) ? r : -r);
        const float t   = (den != 0.f) ? (2.f * apq / den) : 0.f;
        const float c   = 1.f / sqrtf(1.f + t * t);
        const float s   = t * c;
        // Column rotation H <- H*G, and eigenvector accumulation V <- V*G.
#pragma unroll
        for (int k2 = 0; k2 < 8; ++k2) {
          const float hkp = H[k2][p], hkq = H[k2][q];
          H[k2][p] = c * hkp - s * hkq;
          H[k2][q] = s * hkp + c * hkq;
          const float vkp = V[k2][p], vkq = V[k2][q];
          V[k2][p] = c * vkp - s * vkq;
          V[k2][q] = s * vkp + c * vkq;
        }
        // Row rotation H <- G^T * H.
#pragma unroll
        for (int k2 = 0; k2 < 8; ++k2) {
          const float hpk = H[p][k2], hqk = H[q][k2];
          H[p][k2] = c * hpk - s * hqk;
          H[q][k2] = s * hpk + c * hqk;
        }
      }
    }
  }

  // Rank-sort ascending (stable) and scatter into LDS with runtime indices.
  {
    float ev[8];
#pragma unroll
    for (int i = 0; i < 8; ++i) ev[i] = H[i][i];
    int rank[8];
#pragma unroll
    for (int i = 0; i < 8; ++i) {
      int r_ = 0;
#pragma unroll
      for (int j = 0; j < 8; ++j)
        r_ += (ev[j] < ev[i]) || (ev[j] == ev[i] && j < i);
      rank[i] = r_;
    }
#pragma unroll
    for (int i = 0; i < 8; ++i) {
      evS[tid * 8 + rank[i]] = ev[i];
#pragma unroll
      for (int s2 = 0; s2 < 8; ++s2)
        evecS[tid * 64 + s2 * 8 + rank[i]] = V[s2][i];
    }
  }
  __syncthreads();

  // ------------------------- Phase 2: outputs -----------------------------
  // d_out float layout (flattened reference return order):
  const size_t VU_OFF = 0;
  const size_t VV_OFF = (size_t)NMAT * NPAIR * 8;            // 29360128
  const size_t LD_OFF = 2 * VV_OFF;                          // 58720256
  const size_t LU_OFF = LD_OFF + NPAIR;
  const size_t TF_OFF = LU_OFF + NPAIR;
  const size_t EV_OFF = TF_OFF + (size_t)NMAT * NPAIR;       // 62390328

  if (blockIdx.x == 0 && tid < NPAIR) {
    int i0, j0;
    pair_ij(tid, i0, j0);
    out[LD_OFF + tid] = (float)i0;   // lvl_down
    out[LU_OFF + tid] = (float)j0;   // lvl_up
  }

  const int lane = tid & 31;
  const int wave = tid >> 5;
  const int hi   = lane >> 4;

#pragma unroll 1
  for (int it = 0; it < 16; ++it) {
    const int slot0 = wave * 32 + it * 2;            // this wave's matrix pair
    const int g0    = blockIdx.x * MPB + slot0;
    const float fl0 = fl[g0],     fh0 = fh[g0];
    const float fl1 = fl[g0 + 1], fh1 = fh[g0 + 1];

    // A operand: block-diag(evec_m0, evec_m1) as 16x16 in WMMA-A layout,
    // 4 K-slices of 16x4 f32 (2 VGPRs each). Lane holds M = lane&15; reg r
    // holds contraction index kd = 4*(r>>1) + 2*hi + (r&1).
    float a[8];
    {
      const int m  = lane & 15;
      const int bm = m >> 3;
#pragma unroll
      for (int r = 0; r < 8; ++r) {
        const int kd = ((r >> 1) << 2) + (hi << 1) + (r & 1);
        a[r] = ((kd >> 3) == bm)
                   ? evecS[(slot0 + bm) * 64 + (m & 7) * 8 + (kd & 7)]
                   : 0.f;
      }
    }

    // Two 16-column tiles cover the K=28 pair dimension.
#pragma unroll
    for (int ct = 0; ct < 2; ++ct) {
      const int k = ct * 16 + (lane & 15);
      float bu[8], bv[8];
#pragma unroll
      for (int r = 0; r < 8; ++r) {
        const int lrow = ((r >> 1) << 2) + (hi << 1) + (r & 1);
        float vu = 0.f, vv = 0.f;
        if (k < NPAIR) {
          int ik, jk;
          pair_ij(k, ik, jk);
          const int blk = lrow >> 3;
          const float e_i = evS[(slot0 + blk) * 8 + ik];
          const float e_j = evS[(slot0 + blk) * 8 + jk];
          const float tf  = e_j - e_i;
          const float flx = blk ? fl1 : fl0;
          const float fhx = blk ? fh1 : fh0;
          const float vm  = (tf >= flx && tf <= fhx) ? 1.f : 0.f;
          vu = ((lrow & 7) == ik) ? vm : 0.f;   // SelU[l,k]
          vv = ((lrow & 7) == jk) ? vm : 0.f;   // SelV[l,k]
        }
        bu[r] = vu;
        bv[r] = vv;
      }

      // D = A * Sel : chained f32 WMMA over the 16-deep contraction.
      v8f du = {};
      v8f dv = {};
#pragma unroll
      for (int kk = 0; kk < 4; ++kk) {
        v2f av, buv, bvv;
        av.x  = a[2 * kk];      av.y  = a[2 * kk + 1];
        buv.x = bu[2 * kk];     buv.y = bu[2 * kk + 1];
        bvv.x = bv[2 * kk];     bvv.y = bv[2 * kk + 1];
        du = __builtin_amdgcn_wmma_f32_16x16x4_f32(
            false, av, false, buv, (short)0, du, false, false);
        dv = __builtin_amdgcn_wmma_f32_16x16x4_f32(
            false, av, false, bvv, (short)0, dv, false, false);
      }

      // D layout: VGPR v holds (M=v, lanes 0-15) and (M=v+8, lanes 16-31);
      // M = combined s index (matrix block in top bit), N = k tile column.
      if (k < NPAIR) {
#pragma unroll
        for (int v = 0; v < 8; ++v) {
          const int scomb = v + (hi << 3);
          const int bg    = g0 + (scomb >> 3);
          const size_t base =
              (size_t)bg * (NPAIR * 8) + (size_t)k * 8 + (scomb & 7);
          out[VU_OFF + base] = du[v];
          out[VV_OFF + base] = dv[v];
        }
      }
    }

    // transition_freq[b][k] = ev[j]-ev[i], one lane per pair, both matrices.
    if (lane < NPAIR) {
      int i2, j2;
      pair_ij(lane, i2, j2);
      const float tf0 = evS[slot0 * 8 + j2] - evS[slot0 * 8 + i2];
      const float tf1 = evS[(slot0 + 1) * 8 + j2] - evS[(slot0 + 1) * 8 + i2];
      out[TF_OFF + (size_t)g0 * NPAIR + lane]       = tf0;
      out[TF_OFF + (size_t)(g0 + 1) * NPAIR + lane] = tf1;
    }

    // eigen_values[b][k][s] = ev[b][s]: 224 floats/matrix, coalesced b128.
#pragma unroll
    for (int bb = 0; bb < 2; ++bb) {
      const float* evp = &evS[(slot0 + bb) * 8];
      f4 lo4, hi4;
      lo4.x = evp[0]; lo4.y = evp[1]; lo4.z = evp[2]; lo4.w = evp[3];
      hi4.x = evp[4]; hi4.y = evp[5]; hi4.z = evp[6]; hi4.w = evp[7];
      f4 valq;
      if (lane & 1) valq = hi4; else valq = lo4;
      float* base = out + EV_OFF + (size_t)(g0 + bb) * 224;
      *(f4*)(base + lane * 4) = valq;                       // floats [0,128)
      if (lane < 24) *(f4*)(base + 128 + lane * 4) = valq;  // floats [128,224)
    }
  }
}

extern "C" void kernel_launch(void* const* d_in, const int* in_sizes, int n_in,
                              void* d_out, int out_size, void* d_ws,
                              size_t ws_size, hipStream_t stream) {
  const float* F  = (const float*)d_in[0];
  const float* Gz = (const float*)d_in[1];
  const float* fl = (const float*)d_in[2];
  const float* fh = (const float*)d_in[3];
  const float* rf = (const float*)d_in[4];
  (void)in_sizes; (void)n_in; (void)out_size; (void)d_ws; (void)ws_size;

  dim3 grid(NMAT / MPB);   // 2048 blocks, exact cover
  dim3 block(MPB);         // 64 threads = 2 waves (wave32)
  resfreq_kernel<<<grid, block, 0, stream>>>(F, Gz, fl, fh, rf, (float*)d_out);
}